// TokensChooseScatterRouter_24223615549931
// MI455X (gfx1250) — compile-verified
//
#include <hip/hip_runtime.h>
#include <hip/hip_bf16.h>
#include <math.h>

// Problem constants (from reference setup_inputs)
#define G  8
#define T  4096
#define H  2048
#define E  64
#define GT (G * T)          // 32768
#define KSEL 2

typedef float v2f __attribute__((ext_vector_type(2)));
typedef float v8f __attribute__((ext_vector_type(8)));

// ---------------------------------------------------------------------------
// Kernel 1: router GEMM  logits[GT,64] = X[GT,2048] @ W[2048,64] + bias
// One wave per 16-row M tile; block = 8 waves = 128 rows. B staged in LDS
// as (k-pair, n) float2 so each lane does one ds b64 load per N-tile.
// Uses V_WMMA_F32_16X16X4_F32 (exact fp32).
// ---------------------------------------------------------------------------
#define KCHUNK 128

__global__ __launch_bounds__(256)
void router_gemm_kernel(const float* __restrict__ X,
                        const float* __restrict__ W,
                        const float* __restrict__ bias,
                        float* __restrict__ logits)
{
    // pair-packed B: ldsB[kpair][n][0..1] = { W[2*kpair][n], W[2*kpair+1][n] }
    __shared__ float ldsB[KCHUNK / 2][64][2];   // 32 KB

    const int tid   = threadIdx.x;
    const int wave  = tid >> 5;
    const int lane  = tid & 31;
    const int m16   = lane & 15;
    const int khalf = lane >> 4;                 // 0 or 1
    const int rowBase = blockIdx.x * 128 + wave * 16;

    v8f acc0 = {}, acc1 = {}, acc2 = {}, acc3 = {};

    for (int kc = 0; kc < H; kc += KCHUNK) {
        __syncthreads();
        // cooperative stage of B chunk into pair-packed LDS layout
        for (int idx = tid; idx < KCHUNK * 64; idx += 256) {
            int kr = idx >> 6;
            int n  = idx & 63;
            ldsB[kr >> 1][n][kr & 1] = W[(size_t)(kc + kr) * 64 + n];
        }
        __syncthreads();

        const float* Arow = X + (size_t)(rowBase + m16) * H + kc + 2 * khalf;

        #pragma unroll 8
        for (int kk = 0; kk < KCHUNK; kk += 4) {
            // A fragment: this lane holds A[m16][kk+2*khalf], A[m16][kk+2*khalf+1]
            v2f a = *(const v2f*)(Arow + kk);
            int kp = (kk >> 1) + khalf;          // pair index for this lane
            v2f b0 = *(const v2f*)&ldsB[kp][m16 +  0][0];
            v2f b1 = *(const v2f*)&ldsB[kp][m16 + 16][0];
            v2f b2 = *(const v2f*)&ldsB[kp][m16 + 32][0];
            v2f b3 = *(const v2f*)&ldsB[kp][m16 + 48][0];
            acc0 = __builtin_amdgcn_wmma_f32_16x16x4_f32(false, a, false, b0,
                       (short)0, acc0, false, false);
            acc1 = __builtin_amdgcn_wmma_f32_16x16x4_f32(false, a, false, b1,
                       (short)0, acc1, false, false);
            acc2 = __builtin_amdgcn_wmma_f32_16x16x4_f32(false, a, false, b2,
                       (short)0, acc2, false, false);
            acc3 = __builtin_amdgcn_wmma_f32_16x16x4_f32(false, a, false, b3,
                       (short)0, acc3, false, false);
        }
    }

    // C/D layout: VGPR v, lanes 0-15 -> row v, col lane; lanes 16-31 -> row v+8
    const int row0 = rowBase + (khalf ? 8 : 0);
    #pragma unroll
    for (int v = 0; v < 8; ++v) {
        float* out = logits + (size_t)(row0 + v) * 64;
        out[m16 +  0] = acc0[v] + bias[m16 +  0];
        out[m16 + 16] = acc1[v] + bias[m16 + 16];
        out[m16 + 32] = acc2[v] + bias[m16 + 32];
        out[m16 + 48] = acc3[v] + bias[m16 + 48];
    }
}

// ---------------------------------------------------------------------------
// Kernel 2: softmax over E=64, top-2 selection, loss partial sums.
// One thread per token; one block = 256 tokens, all in the same group.
// ---------------------------------------------------------------------------
__global__ __launch_bounds__(256)
void softmax_topk_kernel(const float* __restrict__ logits,
                         float* __restrict__ w1, float* __restrict__ w2,
                         int* __restrict__ idx1, int* __restrict__ idx2,
                         float* __restrict__ probs_sum,   // [G*E]
                         float* __restrict__ mask_sum,    // [G*E]
                         float* __restrict__ z_accum)     // [1]
{
    __shared__ float pprob[E];
    __shared__ float pmask[E];
    __shared__ float zsum;

    const int tid = threadIdx.x;
    if (tid < E) { pprob[tid] = 0.0f; pmask[tid] = 0.0f; }
    if (tid == 0) zsum = 0.0f;
    __syncthreads();

    const int token = blockIdx.x * 256 + tid;   // [0, GT)
    const int g     = token >> 12;              // token / T
    const float* row = logits + (size_t)token * E;

    float m = -INFINITY;
    for (int e = 0; e < E; ++e) m = fmaxf(m, row[e]);
    float s = 0.0f;
    for (int e = 0; e < E; ++e) s += __expf(row[e] - m);
    const float lse = m + __logf(s);
    atomicAdd(&zsum, lse * lse);
    const float inv = 1.0f / s;

    // top-2 (ties -> lowest index, matching lax.top_k)
    float p1 = -1.0f; int i1 = 0;
    for (int e = 0; e < E; ++e) {
        float p = __expf(row[e] - m) * inv;
        if (p > p1) { p1 = p; i1 = e; }
    }
    float p2 = -1.0f; int i2 = 0;
    for (int e = 0; e < E; ++e) {
        if (e == i1) continue;
        float p = __expf(row[e] - m) * inv;
        if (p > p2) { p2 = p; i2 = e; }
    }
    w1[token] = p1; w2[token] = p2;
    idx1[token] = i1; idx2[token] = i2;

    for (int e = 0; e < E; ++e)
        atomicAdd(&pprob[e], __expf(row[e] - m) * inv);
    atomicAdd(&pmask[i1], 1.0f);
    atomicAdd(&pmask[i2], 1.0f);   // i2 != i1 always

    __syncthreads();
    if (tid < E) {
        atomicAdd(&probs_sum[g * E + tid], pprob[tid]);
        atomicAdd(&mask_sum[g * E + tid], pmask[tid]);
    }
    if (tid == 0) atomicAdd(z_accum, zsum);
}

// ---------------------------------------------------------------------------
// Kernel 3: stable descending rank of top-1 prob per group (argsort(-w)).
// rank[i] = #{ j : w[j] > w[i]  ||  (w[j]==w[i] && j < i) }  (stable)
// Also scatters the forward ordering: order[rank] = token.
// 4 blocks of 1024 threads per group; full group's w in LDS (16 KB).
// ---------------------------------------------------------------------------
__global__ __launch_bounds__(1024)
void rank_kernel(const float* __restrict__ w1,
                 int* __restrict__ rank, int* __restrict__ order)
{
    __shared__ float w[T];
    const int g = blockIdx.x >> 2;
    const int quarter = blockIdx.x & 3;
    const int tid = threadIdx.x;

    for (int i = tid; i < T; i += 1024) w[i] = w1[g * T + i];
    __syncthreads();

    const int i = quarter * 1024 + tid;
    const float wi = w[i];
    int cnt = 0;
    for (int j = 0; j < T; ++j) {
        float wj = w[j];
        cnt += (wj > wi) || (wj == wi && j < i);
    }
    rank[g * T + i]    = cnt;
    order[g * T + cnt] = i;
}

// ---------------------------------------------------------------------------
// Kernel 4: capacity positions. Ordered sequence (per group) is
// [top1 choices in rank order (p=0..T-1), top2 choices (p=T..2T-1)].
// priority(item) = count of earlier items with the same expert.
// One wave32 per (g, e): ballot + popc prefix over the 8192 items.
// ---------------------------------------------------------------------------
__global__ __launch_bounds__(32)
void priority_kernel(const int* __restrict__ order,
                     const int* __restrict__ idx1,
                     const int* __restrict__ idx2,
                     int* __restrict__ prio)          // [GT*2]
{
    const int g = blockIdx.x >> 6;
    const int e = blockIdx.x & 63;
    const int lane = threadIdx.x;
    const int* og = order + g * T;
    const int* i1 = idx1 + g * T;
    const int* i2 = idx2 + g * T;

    int cnt = 0;
    for (int base = 0; base < KSEL * T; base += 32) {
        const int p  = base + lane;
        const int kk = p >> 12;        // 0: top1 part, 1: top2 part
        const int r  = p & (T - 1);
        const int token = og[r];
        const int ex = kk ? i2[token] : i1[token];
        const bool match = (ex == e);
        const unsigned mask = __builtin_amdgcn_ballot_w32(match);
        if (match) {
            int pr = cnt + __popc(mask & ((1u << lane) - 1u));
            prio[((size_t)(g * T + token)) * 2 + kk] = pr;
        }
        cnt += __popc(mask);
    }
}

// ---------------------------------------------------------------------------
// Kernel 5: write dispatch_indices (int32) and gated combine weights.
// ---------------------------------------------------------------------------
__global__ __launch_bounds__(256)
void finalize_kernel(const float* __restrict__ w1, const float* __restrict__ w2,
                     const int* __restrict__ idx1, const int* __restrict__ idx2,
                     const int* __restrict__ prio, const int* __restrict__ capPtr,
                     int* __restrict__ disp_out,      // [GT*2*2]
                     float* __restrict__ comb_out)    // [GT*2]
{
    const int token = blockIdx.x * 256 + threadIdx.x;
    const int cap = capPtr[0];
    const int p0 = prio[(size_t)token * 2 + 0];
    const int p1 = prio[(size_t)token * 2 + 1];
    disp_out[(size_t)token * 4 + 0] = idx1[token];
    disp_out[(size_t)token * 4 + 1] = p0;
    disp_out[(size_t)token * 4 + 2] = idx2[token];
    disp_out[(size_t)token * 4 + 3] = p1;
    comb_out[(size_t)token * 2 + 0] = w1[token] * (p0 < cap ? 1.0f : 0.0f);
    comb_out[(size_t)token * 2 + 1] = w2[token] * (p1 < cap ? 1.0f : 0.0f);
}

// ---------------------------------------------------------------------------
// Kernel 6: scalar losses.
// aux = mean_{g,e}( (mask_sum/T) * (probs_sum/T) ) * E^2 ;  z = z_accum/(G*T)
// ---------------------------------------------------------------------------
__global__ void losses_kernel(const float* __restrict__ probs_sum,
                              const float* __restrict__ mask_sum,
                              const float* __restrict__ z_accum,
                              float* __restrict__ out_aux,
                              float* __restrict__ out_z)
{
    if (threadIdx.x == 0 && blockIdx.x == 0) {
        float acc = 0.0f;
        for (int i = 0; i < G * E; ++i)
            acc += (mask_sum[i] * (1.0f / T)) * (probs_sum[i] * (1.0f / T));
        *out_aux = acc * (1.0f / (G * E)) * (float)(E * E);
        *out_z   = z_accum[0] * (1.0f / GT);
    }
}

// ---------------------------------------------------------------------------
extern "C" void kernel_launch(void* const* d_in, const int* in_sizes, int n_in,
                              void* d_out, int out_size, void* d_ws, size_t ws_size,
                              hipStream_t stream)
{
    const float* X    = (const float*)d_in[0];   // [G,T,H]
    const float* W    = (const float*)d_in[1];   // [H,E]
    const float* bias = (const float*)d_in[2];   // [E]
    const int*   cap  = (const int*)d_in[4];     // expert_capacity scalar

    // workspace layout
    char* ws = (char*)d_ws;
    float* logits   = (float*)ws;                              ws += (size_t)GT * E * 4; // 8 MB
    float* w1       = (float*)ws;                              ws += (size_t)GT * 4;
    float* w2       = (float*)ws;                              ws += (size_t)GT * 4;
    int*   idx1     = (int*)ws;                                ws += (size_t)GT * 4;
    int*   idx2     = (int*)ws;                                ws += (size_t)GT * 4;
    int*   rank     = (int*)ws;                                ws += (size_t)GT * 4;
    int*   order    = (int*)ws;                                ws += (size_t)GT * 4;
    int*   prio     = (int*)ws;                                ws += (size_t)GT * 2 * 4;
    float* probsSum = (float*)ws;                              ws += (size_t)G * E * 4;
    float* maskSum  = (float*)ws;                              ws += (size_t)G * E * 4;
    float* zAccum   = (float*)ws;                              ws += 4;

    // zero the accumulators (contiguous tail region), graph-capture safe
    hipMemsetAsync(probsSum, 0, (size_t)(2 * G * E + 1) * 4, stream);

    // output layout: [dispatch int32 GT*2*2][combine f32 GT*2][aux][z]
    int*   disp_out = (int*)d_out;
    float* comb_out = (float*)d_out + (size_t)GT * 4;
    float* aux_out  = (float*)d_out + (size_t)GT * 4 + (size_t)GT * 2;
    float* z_out    = aux_out + 1;

    router_gemm_kernel<<<GT / 128, 256, 0, stream>>>(X, W, bias, logits);
    softmax_topk_kernel<<<GT / 256, 256, 0, stream>>>(logits, w1, w2, idx1, idx2,
                                                      probsSum, maskSum, zAccum);
    rank_kernel<<<G * 4, 1024, 0, stream>>>(w1, rank, order);
    priority_kernel<<<G * E, 32, 0, stream>>>(order, idx1, idx2, prio);
    finalize_kernel<<<GT / 256, 256, 0, stream>>>(w1, w2, idx1, idx2, prio, cap,
                                                  disp_out, comb_out);
    losses_kernel<<<1, 32, 0, stream>>>(probsSum, maskSum, zAccum, aux_out, z_out);
}